// QueryAndGroup_15444702396515
// MI455X (gfx1250) — compile-verified
//
#include <hip/hip_runtime.h>
#include <stdint.h>

typedef __attribute__((ext_vector_type(2))) float v2f;
typedef __attribute__((ext_vector_type(8))) float v8f;

#define B_ 8
#define N_ 8192
#define K_ 2048
#define C_ 64
#define S_ 32
#define RADIUS2 0.0625f
#define INV_R   4.0f

#define CTILE 16                 // centroids per wave (WMMA M dim)
#define NWORDS (N_ / 32)         // 256 mask words per centroid
#define MSTRIDE (NWORDS + 1)     // +1 pad -> no LDS bank conflicts
#define BQ_WAVES 2               // waves per block in ball-query kernel
#define G_WAVES 8                // waves per block in gather kernel

// ---------------------------------------------------------------------------
// Kernel 1: ball query via V_WMMA_F32_16X16X4_F32.
// d2[m][n] = |c_m|^2 - 2 c_m.p_n + |p_n|^2 computed as a single WMMA:
//   A row m = [cx, cy, cz, |c|^2]   (16x4)
//   B col n = [-2px, -2py, -2pz, 1] (4x16)
//   C       = splat(|p_n|^2)        (per lane: column n = lane&15)
// ---------------------------------------------------------------------------
__global__ __launch_bounds__(32 * BQ_WAVES)
void ballquery_wmma_kernel(const float* __restrict__ points,
                           const float* __restrict__ centroids,
                           int* __restrict__ idxbuf)
{
    __shared__ uint32_t lds[BQ_WAVES][CTILE * MSTRIDE];

    const int lid  = threadIdx.x & 31;
    const int wave = threadIdx.x >> 5;
    const int ct   = blockIdx.x * BQ_WAVES + wave;   // centroid-tile id
    const int b    = ct / (K_ / CTILE);
    const int k0   = (ct % (K_ / CTILE)) * CTILE;
    uint32_t* mywords = lds[wave];

    // A fragment: lanes 0-15 hold K=0,1 (cx,cy); lanes 16-31 hold K=2,3 (cz,|c|^2)
    const int m = lid & 15;
    const float* cp = centroids + ((size_t)b * K_ + (k0 + m)) * 3;
    const float cx = cp[0], cy = cp[1], cz = cp[2];
    const float cc = cx * cx + cy * cy + cz * cz;
    v2f afrag;
    afrag.x = (lid < 16) ? cx : cz;
    afrag.y = (lid < 16) ? cy : cc;

    for (int w = 0; w < NWORDS; ++w) {
        uint32_t m0[8], m1[8];
        #pragma unroll
        for (int t = 0; t < 2; ++t) {
            const int n = w * 32 + t * 16 + m;
            const float* pt = points + ((size_t)b * N_ + n) * 3;
            const float px = pt[0], py = pt[1], pz = pt[2];
            const float pp = px * px + py * py + pz * pz;
            v2f bfrag;
            bfrag.x = (lid < 16) ? (-2.0f * px) : (-2.0f * pz);
            bfrag.y = (lid < 16) ? (-2.0f * py) : 1.0f;
            v8f acc = {pp, pp, pp, pp, pp, pp, pp, pp};
            acc = __builtin_amdgcn_wmma_f32_16x16x4_f32(
                      /*neg_a=*/false, afrag, /*neg_b=*/false, bfrag,
                      /*c_mod=*/(short)0, acc, /*reuse_a=*/false, /*reuse_b=*/false);
            uint32_t* mm = (t == 0) ? m0 : m1;
            #pragma unroll
            for (int g = 0; g < 8; ++g)
                mm[g] = (uint32_t)__builtin_amdgcn_ballot_w32(acc[g] < RADIUS2);
        }
        // Assemble per-centroid 32-point mask words; route word m to lane m.
        uint32_t v = 0;
        #pragma unroll
        for (int g = 0; g < 8; ++g) {
            const uint32_t wa = (m0[g] & 0xffffu) | (m1[g] << 16);          // centroid g
            const uint32_t wb = (m0[g] >> 16)     | (m1[g] & 0xffff0000u);  // centroid g+8
            v = (lid == g)     ? wa : v;
            v = (lid == g + 8) ? wb : v;
        }
        if (lid < 16) mywords[lid * MSTRIDE + w] = v;
    }
    __syncthreads();

    // Selection: lane m scans centroid m's bitmap in point-index order.
    if (lid < 16) {
        int* out = idxbuf + ((size_t)b * K_ + (k0 + lid)) * S_;
        int cnt = 0, first = 0;
        for (int w = 0; w < NWORDS && cnt < S_; ++w) {
            uint32_t word = mywords[lid * MSTRIDE + w];
            while (word && cnt < S_) {
                const int bit = __builtin_ctz(word);
                word &= word - 1;
                const int n = w * 32 + bit;
                if (cnt == 0) first = n;
                out[cnt++] = n;
            }
        }
        for (int s = cnt; s < S_; ++s) out[s] = first;  // pad rule (0 if empty)
    }
}

// ---------------------------------------------------------------------------
// Kernel 2: gather + normalize. One wave per centroid; lane = sample slot.
// Stores are fully coalesced (32 consecutive floats per wave per row).
// ---------------------------------------------------------------------------
__global__ __launch_bounds__(32 * G_WAVES)
void group_gather_kernel(const float* __restrict__ points,
                         const float* __restrict__ centroids,
                         const float* __restrict__ features,
                         const int* __restrict__ idxbuf,
                         float* __restrict__ out)
{
    const int lid  = threadIdx.x & 31;
    const int wave = threadIdx.x >> 5;
    const int gc   = blockIdx.x * G_WAVES + wave;   // global centroid id
    const int b = gc / K_;
    const int k = gc % K_;

    const int i = idxbuf[((size_t)b * K_ + k) * S_ + lid];

    const float* cp = centroids + ((size_t)b * K_ + k) * 3;
    const float cx = cp[0], cy = cp[1], cz = cp[2];
    const float* pt = points + ((size_t)b * N_ + i) * 3;

    float* gout = out;  // grouped_pts: (B,3,K,S)
    gout[(((size_t)b * 3 + 0) * K_ + k) * S_ + lid] = (pt[0] - cx) * INV_R;
    gout[(((size_t)b * 3 + 1) * K_ + k) * S_ + lid] = (pt[1] - cy) * INV_R;
    gout[(((size_t)b * 3 + 2) * K_ + k) * S_ + lid] = (pt[2] - cz) * INV_R;

    float* fout = out + (size_t)B_ * 3 * K_ * S_;   // new_feats: (B,C,K,S)
    const float* fbase = features + (size_t)b * C_ * N_;
    #pragma unroll 4
    for (int ch = 0; ch < C_; ++ch) {
        const float v = fbase[(size_t)ch * N_ + i];
        fout[(((size_t)b * C_ + ch) * K_ + k) * S_ + lid] = v;
    }
}

// ---------------------------------------------------------------------------
extern "C" void kernel_launch(void* const* d_in, const int* in_sizes, int n_in,
                              void* d_out, int out_size, void* d_ws, size_t ws_size,
                              hipStream_t stream)
{
    const float* points    = (const float*)d_in[0];   // (B, N, 3)
    const float* centroids = (const float*)d_in[1];   // (B, K, 3)
    const float* features  = (const float*)d_in[2];   // (B, C, N)
    float* out  = (float*)d_out;
    int* idxbuf = (int*)d_ws;                         // B*K*S ints = 2 MB

    const int bq_blocks = (B_ * K_ / CTILE) / BQ_WAVES;  // 1024 waves -> 512 blocks
    ballquery_wmma_kernel<<<bq_blocks, 32 * BQ_WAVES, 0, stream>>>(
        points, centroids, idxbuf);

    const int g_blocks = (B_ * K_) / G_WAVES;            // 16384 waves -> 2048 blocks
    group_gather_kernel<<<g_blocks, 32 * G_WAVES, 0, stream>>>(
        points, centroids, features, idxbuf, out);
}